// DSG_STGCN_7567732376315
// MI455X (gfx1250) — compile-verified
//
#include <hip/hip_runtime.h>

typedef __attribute__((ext_vector_type(2))) float v2f;
typedef __attribute__((ext_vector_type(8))) float v8f;

#define B_  64
#define N_  128
#define T_  128
#define V_  2052
#define K_  512
#define H_  64
#define VPAD 2080   // pad to multiple of 32 for 32x32 macro-tiles
#define BT  8192    // B_*T_

__device__ __forceinline__ v8f wmma_f32(v2f a, v2f b, v8f c) {
  // V_WMMA_F32_16X16X4_F32 : D = A(16x4) * B(4x16) + C(16x16)
  return __builtin_amdgcn_wmma_f32_16x16x4_f32(
      false, a, false, b, (short)0, c, false, false);
}

// f32 A/B fragment load per ISA layout:
// lanes 0-15  : row = row0+lane,    K = kb+0, kb+1
// lanes 16-31 : row = row0+lane-16, K = kb+2, kb+3
__device__ __forceinline__ v2f load_frag(const float* __restrict__ base,
                                         int row0, int ldk, int kb, int lane) {
  int half = lane >> 4;
  int r = lane & 15;
  const float* p = base + (size_t)(row0 + r) * ldk + kb + half * 2;
  return *(const v2f*)p;
}

// ---------------- Hilbert circular-convolution kernel taps -----------------
__global__ void k_kvec(float* __restrict__ kre, float* __restrict__ kim) {
  int d = threadIdx.x;                       // 128 threads
  float sr = 0.f, si = 0.f;
  for (int k = 0; k <= 64; ++k) {
    float hf = (k == 0 || k == 64) ? 1.0f : 2.0f;
    float ang = 6.283185307179586f * (float)k * (float)d / 128.0f;
    sr += hf * __cosf(ang);
    si += hf * __sinf(ang);
  }
  kre[d] = sr * (1.0f / 128.0f);
  kim[d] = si * (1.0f / 128.0f);
}

// ---------------- analytic signal -> unit phasors (C=Re, S=Im) -------------
// C,S stored electrode-major: C[n, b*T+t]  (K-contiguous for the Gram GEMM)
__global__ void k_analytic(const float* __restrict__ z,
                           const float* __restrict__ kre,
                           const float* __restrict__ kim,
                           float* __restrict__ Cm, float* __restrict__ Sm) {
  __shared__ float skre[128], skim[128], sz[2][128];
  int t    = threadIdx.x & 127;
  int half = threadIdx.x >> 7;
  int row  = blockIdx.x * 2 + half;          // row = b*N_ + n, in [0, 8192)
  if (half == 0) skre[t] = kre[t]; else skim[t] = kim[t];
  sz[half][t] = z[(size_t)row * T_ + t];
  __syncthreads();
  float ar = 0.f, ai = 0.f;
  for (int tt = 0; tt < 128; ++tt) {
    int d = (t - tt) & 127;
    float zv = sz[half][tt];
    ar += skre[d] * zv;
    ai += skim[d] * zv;
  }
  float mag = sqrtf(ar * ar + ai * ai);
  float inv = (mag > 1e-30f) ? 1.0f / mag : 0.0f;
  float c   = (mag > 1e-30f) ? ar * inv : 1.0f;
  float s   = ai * inv;
  int b = row >> 7, n = row & 127;
  Cm[(size_t)n * BT + b * T_ + t] = c;
  Sm[(size_t)n * BT + b * T_ + t] = s;
}

// ---------------- PLV Gram via fp32 WMMA, fused threshold ------------------
// G = E E^H : Re = C C^T + S S^T, Im = S C^T - C S^T ; a = (|G|/8192 >= 0.5)
__global__ void k_gram(const float* __restrict__ C, const float* __restrict__ S,
                       float* __restrict__ a_out) {
  int wid  = blockIdx.x * (blockDim.x >> 5) + (threadIdx.x >> 5);
  int lane = threadIdx.x & 31;
  int jt = wid & 7, it = wid >> 3;           // 8x8 tiles of 16x16
  v8f accR = {}, accI = {};
  for (int kk = 0; kk < BT / 4; ++kk) {
    int kb = kk * 4;
    v2f aC = load_frag(C, it * 16, BT, kb, lane);
    v2f aS = load_frag(S, it * 16, BT, kb, lane);
    v2f bC = load_frag(C, jt * 16, BT, kb, lane);
    v2f bS = load_frag(S, jt * 16, BT, kb, lane);
    accR = wmma_f32(aC, bC, accR);
    accR = wmma_f32(aS, bS, accR);
    accI = wmma_f32(aS, bC, accI);
    accI = wmma_f32(-aC, bS, accI);
  }
  int col   = jt * 16 + (lane & 15);
  int rbase = it * 16 + ((lane >> 4) << 3);
#pragma unroll
  for (int r = 0; r < 8; ++r) {
    float gr = accR[r], gi = accI[r];
    float plv = sqrtf(gr * gr + gi * gi) * (1.0f / (float)BT);
    a_out[(size_t)(rbase + r) * N_ + col] = (plv >= 0.5f) ? 1.0f : 0.0f;
  }
}

// ---------------- DCT matrix (512 x 2052) ----------------------------------
__global__ void k_dct(float* __restrict__ dct) {
  int tid = blockIdx.x * blockDim.x + threadIdx.x;
  if (tid >= K_ * V_) return;
  int k = tid / V_, v = tid % V_;
  float val;
  if (k == 0) val = rsqrtf((float)V_);
  else val = sqrtf(2.0f / (float)V_) *
             cosf(3.14159265358979323846f * (float)(2 * v + 1) * (float)k /
                  (2.0f * (float)V_));
  dct[tid] = val;
}

// M = dct @ LF : (512 x 128)
__global__ void k_M(const float* __restrict__ dct, const float* __restrict__ LF,
                    float* __restrict__ M) {
  int tid = blockIdx.x * blockDim.x + threadIdx.x;  // 512*128
  int k = tid >> 7, e = tid & 127;
  float acc = 0.f;
  for (int v = 0; v < V_; ++v)
    acc += dct[(size_t)k * V_ + v] * LF[(size_t)v * N_ + e];
  M[tid] = acc;
}

// Rpad = dct^T @ M : (2080 x 128), rows >= 2052 zero-padded
__global__ void k_R(const float* __restrict__ dct, const float* __restrict__ M,
                    float* __restrict__ Rp) {
  int tid = blockIdx.x * blockDim.x + threadIdx.x;  // 2080*128
  int v = tid >> 7, e = tid & 127;
  float acc = 0.f;
  if (v < V_)
    for (int k = 0; k < K_; ++k)
      acc += dct[(size_t)k * V_ + v] * M[(size_t)k * N_ + e];
  Rp[tid] = acc;
}

// ---------------- batched GEMM out[b] = A @ z[b]^T via fp32 WMMA -----------
// out[b, m, n] = sum_t A[m, t] * z[b, n, t]
// One wave computes a 32x32 macro-tile: 4 accumulators, each A/B fragment
// reused twice -> 4 loads : 4 WMMAs per K-step.
__global__ void k_gemm(const float* __restrict__ A, const float* __restrict__ z,
                       float* __restrict__ out, int mtiles2, int mvalid,
                       long bstride) {
  int wid  = blockIdx.x * (blockDim.x >> 5) + (threadIdx.x >> 5);
  int lane = threadIdx.x & 31;
  int nt = wid & 3;                      // 4 macro col-tiles of 32 (N=128)
  int mt = (wid >> 2) % mtiles2;         // macro row-tile of 32
  int b  = wid / (mtiles2 << 2);
  const float* zb = z + (size_t)b * (N_ * T_);
  int m0 = mt * 32, n0 = nt * 32;
  v8f acc00 = {}, acc01 = {}, acc10 = {}, acc11 = {};
  for (int kk = 0; kk < T_ / 4; ++kk) {
    int kb = kk * 4;
    v2f a0 = load_frag(A, m0, T_, kb, lane);
    v2f a1 = load_frag(A, m0 + 16, T_, kb, lane);
    v2f b0 = load_frag(zb, n0, T_, kb, lane);       // B[k][n] = z[b][n][k]
    v2f b1 = load_frag(zb, n0 + 16, T_, kb, lane);
    acc00 = wmma_f32(a0, b0, acc00);
    acc01 = wmma_f32(a0, b1, acc01);
    acc10 = wmma_f32(a1, b0, acc10);
    acc11 = wmma_f32(a1, b1, acc11);
  }
  float* ob = out + (size_t)b * bstride;
  int colb = n0 + (lane & 15);
  int rowb = m0 + ((lane >> 4) << 3);
#pragma unroll
  for (int r = 0; r < 8; ++r) {
    int r0 = rowb + r, r1 = rowb + 16 + r;
    if (r0 < mvalid) {
      __builtin_nontemporal_store(acc00[r], &ob[(size_t)r0 * N_ + colb]);
      __builtin_nontemporal_store(acc01[r], &ob[(size_t)r0 * N_ + colb + 16]);
    }
    if (r1 < mvalid) {
      __builtin_nontemporal_store(acc10[r], &ob[(size_t)r1 * N_ + colb]);
      __builtin_nontemporal_store(acc11[r], &ob[(size_t)r1 * N_ + colb + 16]);
    }
  }
}

// ---------------- small GAE / gumbel chain ---------------------------------
__global__ void k_x(const float* __restrict__ z, float* __restrict__ x) {
  int tid = blockIdx.x * blockDim.x + threadIdx.x;  // 16384
  float acc = 0.f;
  for (int b = 0; b < B_; ++b) acc += z[(size_t)b * (N_ * T_) + tid];
  x[tid] = acc * (1.0f / (float)B_);
}

__global__ void k_ax(const float* __restrict__ a, const float* __restrict__ x,
                     float* __restrict__ ax) {
  int tid = blockIdx.x * blockDim.x + threadIdx.x;  // 16384
  int i = tid >> 7, t = tid & 127;
  float acc = 0.f;
  for (int j = 0; j < N_; ++j) acc += a[i * N_ + j] * x[j * T_ + t];
  ax[tid] = acc;
}

__global__ void k_hg(const float* __restrict__ ax, const float* __restrict__ wg,
                     const float* __restrict__ bg, float* __restrict__ hg) {
  int tid = blockIdx.x * blockDim.x + threadIdx.x;  // 128*64
  int i = tid >> 6, h = tid & 63;
  float acc = bg[h];
  for (int t = 0; t < T_; ++t) acc += ax[i * T_ + t] * wg[t * H_ + h];
  hg[tid] = acc > 0.f ? acc : 0.f;
}

__global__ void k_aaug(const float* __restrict__ hg, const float* __restrict__ a,
                       const float* __restrict__ u, float* __restrict__ out) {
  int tid = blockIdx.x * blockDim.x + threadIdx.x;  // 16384
  int i = tid >> 7, j = tid & 127;
  float dot = 0.f;
  for (int h = 0; h < H_; ++h) dot += hg[i * H_ + h] * hg[j * H_ + h];
  float p = 1.0f / (1.0f + expf(-dot));
  float e = 0.5f * p + 0.5f * a[tid];                 // RHO = 0.5
  float logit = logf(e) - logf(1.0f - e);
  float g = -logf(-logf(u[tid]));
  float zarg = (logit + g) * 10.0f;                   // / TAU (0.1)
  out[tid] = 1.0f / (1.0f + expf(-zarg));
}

extern "C" void kernel_launch(void* const* d_in, const int* in_sizes, int n_in,
                              void* d_out, int out_size, void* d_ws, size_t ws_size,
                              hipStream_t stream) {
  const float* z   = (const float*)d_in[0];
  const float* LF  = (const float*)d_in[1];
  const float* gu  = (const float*)d_in[2];
  const float* wg  = (const float*)d_in[3];
  const float* bg  = (const float*)d_in[4];
  // d_in[5..12] (w1,b1,w2,b2,w_ih,w_hh,b_ih,b_hh) feed only the GRU/GCN path
  // whose results are discarded by the reference -> dead code, unused here.
  float* out = (float*)d_out;
  float* ws  = (float*)d_ws;

  // workspace layout (floats)
  float* kre = ws;                       // 128
  float* kim = kre + 128;                // 128
  float* Cm  = kim + 128;                // 128*8192
  float* Sm  = Cm + (size_t)N_ * BT;     // 128*8192
  float* a_  = Sm + (size_t)N_ * BT;     // 16384
  float* x_  = a_ + N_ * N_;             // 16384
  float* ax_ = x_ + N_ * T_;             // 16384
  float* hg_ = ax_ + N_ * T_;            // 8192
  float* dct = hg_ + N_ * H_;            // 512*2052
  float* M_  = dct + (size_t)K_ * V_;    // 512*128
  float* Rp  = M_ + (size_t)K_ * N_;     // 2080*128

  float* out_slow  = out;                                   // 64*512*128
  float* out_srec  = out + (size_t)B_ * K_ * N_;            // 64*2052*128
  float* out_aaug  = out_srec + (size_t)B_ * V_ * N_;       // 128*128

  // --- phase/PLV/adjacency chain ---
  k_kvec<<<1, 128, 0, stream>>>(kre, kim);
  k_analytic<<<(B_ * N_) / 2, 256, 0, stream>>>(z, kre, kim, Cm, Sm);
  k_gram<<<16, 128, 0, stream>>>(Cm, Sm, a_);               // 64 waves, WMMA

  // --- DCT-fused projection matrices ---
  k_dct<<<(K_ * V_ + 255) / 256, 256, 0, stream>>>(dct);
  k_M<<<(K_ * N_) / 256, 256, 0, stream>>>(dct, LF, M_);
  k_R<<<(VPAD * N_) / 256, 256, 0, stream>>>(dct, M_, Rp);

  // --- tiny GAE / gumbel chain -> a_aug output ---
  k_x<<<64, 256, 0, stream>>>(z, x_);
  k_ax<<<64, 256, 0, stream>>>(a_, x_, ax_);
  k_hg<<<32, 256, 0, stream>>>(ax_, wg, bg, hg_);
  k_aaug<<<64, 256, 0, stream>>>(hg_, a_, gu, out_aaug);

  // --- big WMMA GEMMs: s_low, s_recon (32x32 macro-tiles per wave) ---
  // s_low : 64 * (512/32=16) * (128/32=4) = 4096 waves, 4 waves/block
  k_gemm<<<(B_ * 16 * 4) / 4, 128, 0, stream>>>(M_, z, out_slow, 16, K_,
                                                (long)K_ * N_);
  // s_recon : 64 * (2080/32=65) * 4 = 16640 waves
  k_gemm<<<(B_ * 65 * 4) / 4, 128, 0, stream>>>(Rp, z, out_srec, 65, V_,
                                                (long)V_ * N_);
}